// TVL1OF_29489245454497
// MI455X (gfx1250) — compile-verified
//
#include <hip/hip_runtime.h>
#include <hip/hip_bf16.h>

// TV-L1 optical flow, MI455X / gfx1250, wave32.
// - Sobel 3x3 convs mapped to V_WMMA_F32_16X16X4_F32 patch-GEMMs.
// - Conv input tiles staged into LDS by the Tensor Data Mover
//   (tensor_load_to_lds + s_wait_tensorcnt) for interior tiles.
// - Everything else is L2-resident pointwise streaming (~48 MB working set
//   vs 192 MB L2).

typedef __attribute__((ext_vector_type(2))) float v2f;
typedef __attribute__((ext_vector_type(8))) float v8f;
typedef __attribute__((ext_vector_type(4))) unsigned int v4u32;
typedef __attribute__((ext_vector_type(8))) int v8i32;
typedef __attribute__((ext_vector_type(4))) int v4i32;

#define Wd 1024
#define Hd 1024
#define NPIX (Wd * Hd)

#ifndef __has_builtin
#define __has_builtin(x) 0
#endif
#if defined(__AMDGCN__) && __has_builtin(__builtin_amdgcn_tensor_load_to_lds) && \
    __has_builtin(__builtin_amdgcn_s_wait_tensorcnt)
#define HAVE_TDM 1
#else
#define HAVE_TDM 0
#endif

// ---------------------------------------------------------------------------
// Sobel conv via WMMA. One 256-thread block = 8 waves = one 8-row x 16-col
// output tile. Each wave computes one 16-pixel strip x up to 4 output maps
// as D(16x16) = A(16xK) * B(Kx16) with K = 9*CH (padded to 4*NCHUNK).
//   CH==1: in = single image plane -> out maps {gx, gy} (+ optional norm)
//   CH==2: in = u (2 planes)      -> out maps {gx(u0),gy(u0),gx(u1),gy(u1)}
//          (+ global sum of abs via LDS atomic + global atomic)
// A layout (16x4 f32): lanes 0-15 hold M=lane, VGPR0/1 = K0/K1;
//                      lanes 16-31 hold M=lane-16, VGPR0/1 = K2/K3.
// B layout (4x16 f32): lane l, VGPR j -> K = 2*(l/16)+j, N = l%16.
// D layout (16x16 f32): lane l, VGPR r -> N = l%16, M = r + 8*(l/16).
// ---------------------------------------------------------------------------
template <int CH>
__global__ __launch_bounds__(256) void sobel_wmma_kernel(
    const float* __restrict__ in, const float* __restrict__ gw,
    float* __restrict__ out, float* __restrict__ norm_out,
    float* __restrict__ sum_accum) {
  constexpr int NCHUNK = (CH == 1) ? 3 : 5;  // ceil(9*CH / 4)
  const int tid = threadIdx.x;
  const int tileX = blockIdx.x;  // Wd/16 tiles
  const int tileY = blockIdx.y;  // Hd/8 tiles

  // Contiguous rows (18 floats) so the TDM can write them back-to-back.
  __shared__ float sIn[CH][10][18];
  __shared__ float sSum;
  if (tid == 0) sSum = 0.0f;

  bool tile_loaded = false;
#if HAVE_TDM
  // Interior tiles: one 3D TDM transfer stages the whole (x=18, y=10, z=CH)
  // halo tile from the strided image planes straight into LDS. Border tiles
  // need zero padding, which the descriptor cannot express -> fallback.
  const bool interior = (tileX > 0) && (tileX < (Wd / 16) - 1) &&
                        (tileY > 0) && (tileY < (Hd / 8) - 1);
  if (interior) {
    tile_loaded = true;
    if (tid < 32) {  // one wave issues the DMA; TENSORcnt is per-wave
      const unsigned long long ga =
          (unsigned long long)(uintptr_t)in +
          4ull * ((unsigned long long)(tileY * 8 - 1) * Wd + (tileX * 16 - 1));
      const unsigned int lds_addr = (unsigned int)(uintptr_t)(&sIn[0][0][0]);

      // D# group 0: count=1 | lds_addr | global_addr[56:0] | type=2.
      v4u32 g0;
      g0[0] = 1u;
      g0[1] = lds_addr;
      g0[2] = (unsigned int)ga;
      g0[3] = ((unsigned int)(ga >> 32) & 0x01FFFFFFu) | 0x80000000u;

      // D# group 1: data_size=4B; tensor 1024x1024 (xNPIX stride for z);
      // tile 18 x 10 x CH.
      v8i32 g1;
      g1[0] = (int)(2u << 16);              // data_size=2 (4 bytes)
      g1[1] = (int)(1024u << 16);           // tensor_dim0[15:0] @ [31:16]
      g1[2] = (int)(1024u << 16);           // td0[31:16]=0 | tensor_dim1[15:0]
      g1[3] = (int)(18u << 16);             // td1[31:16]=0 | tile_dim0=18
      g1[4] = (int)(((unsigned)CH << 16) | 10u);  // tile_dim1=10, tile_dim2=CH
      g1[5] = 1024;                         // tensor_dim0_stride[31:0]
      g1[6] = 0;                            // stride0[47:32] | stride1[15:0]
      g1[7] = (int)((unsigned long long)NPIX >> 16);  // stride1[47:16]

      // D# group 2: tensor_dim2=CH; no dim3; group 3: zeros.
      v4i32 g2;
      g2[0] = CH; g2[1] = 0; g2[2] = 0; g2[3] = 0;
      v4i32 g3;
      g3[0] = 0; g3[1] = 0; g3[2] = 0; g3[3] = 0;

#if __clang_major__ >= 23
      v8i32 g4;
      g4[0] = 0; g4[1] = 0; g4[2] = 0; g4[3] = 0;
      g4[4] = 0; g4[5] = 0; g4[6] = 0; g4[7] = 0;
      __builtin_amdgcn_tensor_load_to_lds(g0, g1, g2, g3, g4, 0);
#else
      __builtin_amdgcn_tensor_load_to_lds(g0, g1, g2, g3, 0);
#endif
      __builtin_amdgcn_s_wait_tensorcnt(0);
    }
  }
#endif  // HAVE_TDM

  if (!tile_loaded) {
    // Cooperative zero-padded staging (border tiles / no-TDM toolchain).
    const int tot = CH * 10 * 18;
    for (int e = tid; e < tot; e += 256) {
      const int ch = e / 180;
      const int rem = e % 180;
      const int ry = rem / 18;
      const int rx = rem % 18;
      const int gy = tileY * 8 + ry - 1;
      const int gx = tileX * 16 + rx - 1;
      float v = 0.0f;
      if (gy >= 0 && gy < Hd && gx >= 0 && gx < Wd)
        v = in[(size_t)ch * NPIX + (size_t)gy * Wd + gx];
      sIn[ch][ry][rx] = v;
    }
  }
  __syncthreads();

  const int lane = tid & 31;
  const int wave = tid >> 5;  // output row within tile (0..7)
  const int half = lane >> 4;
  const int m = lane & 15;  // A: pixel (col in strip); B/D: N (output map)

  // Build constant block-diagonal weight matrix B in registers.
  v2f b[NCHUNK];
#pragma unroll
  for (int q = 0; q < NCHUNK; ++q) {
#pragma unroll
    for (int j = 0; j < 2; ++j) {
      const int k = 4 * q + 2 * half + j;
      float bv = 0.0f;
      if (m < 2 * CH && k < 9 * CH) {
        const int ch = k / 9;
        const int t = k % 9;
        const int ic = (CH == 2) ? (m >> 1) : 0;  // input channel of map m
        const int f = (CH == 2) ? (m & 1) : m;    // filter (0=gx, 1=gy)
        if (ch == ic) bv = gw[f * 9 + t];
      }
      b[q][j] = bv;
    }
  }

  // Patch-GEMM: accumulate over K chunks.
  v8f c = {};
#pragma unroll
  for (int q = 0; q < NCHUNK; ++q) {
    v2f a;
#pragma unroll
    for (int j = 0; j < 2; ++j) {
      const int k = 4 * q + 2 * half + j;
      float av = 0.0f;
      if (k < 9 * CH) {
        const int ch = k / 9;
        const int t = k % 9;
        av = sIn[ch][wave + t / 3][m + (t % 3)];
      }
      a[j] = av;
    }
    c = __builtin_amdgcn_wmma_f32_16x16x4_f32(false, a, false, b[q], (short)0,
                                              c, false, false);
  }

  const int y = tileY * 8 + wave;
  const int x0 = tileX * 16 + 8 * half;  // this lane holds cols x0..x0+7

  // Optional norm = gx^2 + gy^2 (init pass): gy lives in lane N=1 of the
  // same half for the same pixels; pull it over with a wave shuffle.
  if (CH == 1) {
    float nv[8];
#pragma unroll
    for (int r = 0; r < 8; ++r) {
      const float gyv = __shfl(c[r], (half << 4) + 1, 32);
      nv[r] = c[r] * c[r] + gyv * gyv;
    }
    if (m == 0 && norm_out) {
      float* np = norm_out + (size_t)y * Wd + x0;
      *(float4*)np = make_float4(nv[0], nv[1], nv[2], nv[3]);
      *(float4*)(np + 4) = make_float4(nv[4], nv[5], nv[6], nv[7]);
    }
  }

  if (m < 2 * CH) {
    float* op = out + (size_t)m * NPIX + (size_t)y * Wd + x0;
    *(float4*)op = make_float4(c[0], c[1], c[2], c[3]);
    *(float4*)(op + 4) = make_float4(c[4], c[5], c[6], c[7]);
    if (sum_accum) {
      const float s = fabsf(c[0]) + fabsf(c[1]) + fabsf(c[2]) + fabsf(c[3]) +
                      fabsf(c[4]) + fabsf(c[5]) + fabsf(c[6]) + fabsf(c[7]);
      atomicAdd(&sSum, s);  // ds_add_f32
    }
  }

  __syncthreads();
  if (sum_accum && tid == 0) atomicAdd(sum_accum, sSum);  // global_atomic_add_f32
}

// ---------------------------------------------------------------------------
// Init: rho_c = x1 - x0, u = 0, p = 0. Grid covers 4*NPIX.
// ---------------------------------------------------------------------------
__global__ __launch_bounds__(256) void init_kernel(const float* __restrict__ x,
                                                   float* __restrict__ rhoc,
                                                   float* __restrict__ u,
                                                   float* __restrict__ p) {
  const int idx = blockIdx.x * 256 + threadIdx.x;
  if (idx < NPIX) rhoc[idx] = x[NPIX + idx] - x[idx];
  if (idx < 2 * NPIX) u[idx] = 0.0f;
  if (idx < 4 * NPIX) p[idx] = 0.0f;
}

// ---------------------------------------------------------------------------
// u-update: pointwise in u, small stencil in p. Also zeroes the abs-sum
// accumulator for this iteration (runs strictly before the gradu pass).
// ---------------------------------------------------------------------------
__global__ __launch_bounds__(256) void uupdate_kernel(
    const float* __restrict__ gradim, const float* __restrict__ norm,
    const float* __restrict__ rhoc, float* __restrict__ u,
    const float* __restrict__ p, const float* __restrict__ wxp,
    const float* __restrict__ wyp, const float* __restrict__ lamp,
    const float* __restrict__ taup, const float* __restrict__ thetap,
    float* __restrict__ sum_accum) {
  const int idx = blockIdx.x * 256 + threadIdx.x;
  if (idx == 0) *sum_accum = 0.0f;
  if (idx >= NPIX) return;
  const int y = idx / Wd;
  const int x = idx - y * Wd;

  const float theta = *thetap;
  const float tl = theta * (*lamp);

  const float g0 = gradim[idx];
  const float g1 = gradim[NPIX + idx];
  const float ng = norm[idx];
  const float u0 = u[idx];
  const float u1 = u[NPIX + idx];

  const float rho = rhoc[idx] + g0 * u0 + g1 * u1;
  const float th = tl * ng;
  const float a = fabsf(rho);
  float coef = 0.0f;
  if (a < th)
    coef = rho / ng;
  else if (a > th)
    coef = tl * (float)((rho > 0.0f) - (rho < 0.0f));
  const float v0 = u0 - coef * g0;
  const float v1 = u1 - coef * g1;

  const float wx0 = wxp[0], wx1 = wxp[1];
  const float wy0 = wyp[0], wy1 = wyp[1];
  const float p00 = p[idx];
  const float p01 = p[NPIX + idx];
  const float p10 = p[2 * NPIX + idx];
  const float p11 = p[3 * NPIX + idx];
  const float p00l = (x > 0) ? p[idx - 1] : 0.0f;
  const float p01l = (x > 0) ? p[NPIX + idx - 1] : 0.0f;
  const float p10u = (y > 0) ? p[2 * NPIX + idx - Wd] : 0.0f;
  const float p11u = (y > 0) ? p[3 * NPIX + idx - Wd] : 0.0f;

  const float div0 = (p00l * wx0 + p00 * wx1) + (p10u * wy0 + p10 * wy1);
  const float div1 = (p01l * wx0 + p01 * wx1) + (p11u * wy0 + p11 * wy1);

  u[idx] = v0 + theta * div0;
  u[NPIX + idx] = v1 + theta * div1;
}

// ---------------------------------------------------------------------------
// p-update: p = (p + r*gradu) / (1 + r * sum|gradu|), flat over 4*NPIX.
// ---------------------------------------------------------------------------
__global__ __launch_bounds__(256) void pupdate_kernel(
    float* __restrict__ p, const float* __restrict__ gradu,
    const float* __restrict__ sum_accum, const float* __restrict__ taup,
    const float* __restrict__ thetap) {
  const int idx = blockIdx.x * 256 + threadIdx.x;
  if (idx >= 4 * NPIX) return;
  const float r = (*taup) / (*thetap);
  const float inv = 1.0f / (1.0f + r * (*sum_accum));
  p[idx] = (p[idx] + r * gradu[idx]) * inv;
}

// ---------------------------------------------------------------------------
extern "C" void kernel_launch(void* const* d_in, const int* in_sizes, int n_in,
                              void* d_out, int out_size, void* d_ws,
                              size_t ws_size, hipStream_t stream) {
  const float* x = (const float*)d_in[0];      // (1,2,H,W)
  const float* gw = (const float*)d_in[1];     // (2,1,3,3) flat 18
  const float* wx = (const float*)d_in[2];     // (1,2)
  const float* wy = (const float*)d_in[3];     // (2,1)
  const float* lam = (const float*)d_in[4];
  const float* tau = (const float*)d_in[5];
  const float* theta = (const float*)d_in[6];
  float* u = (float*)d_out;  // (1,2,H,W)

  // Workspace layout (floats): grad_im[2N] | norm[N] | rho_c[N] | p[4N] |
  // gradu[4N] | sum[1]  -> ~48 MB, L2-resident on MI455X (192 MB).
  float* ws = (float*)d_ws;
  float* gradim = ws;
  float* norm = ws + (size_t)2 * NPIX;
  float* rhoc = ws + (size_t)3 * NPIX;
  float* p = ws + (size_t)4 * NPIX;
  float* gradu = ws + (size_t)8 * NPIX;
  float* sum = ws + (size_t)12 * NPIX;

  const dim3 convGrid(Wd / 16, Hd / 8);

  init_kernel<<<(4 * NPIX) / 256, 256, 0, stream>>>(x, rhoc, u, p);
  sobel_wmma_kernel<1><<<convGrid, 256, 0, stream>>>(x + NPIX, gw, gradim,
                                                     norm, nullptr);

  for (int it = 0; it < 10; ++it) {
    uupdate_kernel<<<NPIX / 256, 256, 0, stream>>>(gradim, norm, rhoc, u, p,
                                                   wx, wy, lam, tau, theta,
                                                   sum);
    sobel_wmma_kernel<2><<<convGrid, 256, 0, stream>>>(u, gw, gradu, nullptr,
                                                       sum);
    pupdate_kernel<<<(4 * NPIX) / 256, 256, 0, stream>>>(p, gradu, sum, tau,
                                                         theta);
  }
}